// SoftMaskingModule_48395691491576
// MI455X (gfx1250) — compile-verified
//
#include <hip/hip_runtime.h>
#include <cstddef>
#include <cstdint>

#define TPB 256
constexpr int kV     = 50257;
constexpr int kH     = 1024;
constexpr int kK     = 5;
constexpr int kMask  = 50256;
constexpr int kChunk = TPB * 4;                 // 1024 floats per block-iteration
constexpr int kIters = kV / kChunk;             // 49
constexpr int kTail  = kV - kIters * kChunk;    // 81
constexpr int kDepth = 4;                       // async pipeline depth (ring buffers)

// ---- CDNA5 async global->LDS copy (ASYNCcnt-tracked DMA-style prefetch) ----
__device__ __forceinline__ void async_copy_b128(const float* gsrc, void* lds_dst) {
  // low 32 bits of a generic LDS pointer == LDS byte offset (ISA 10.2 aperture rules)
  unsigned loff = (unsigned)(size_t)lds_dst;
  asm volatile("global_load_async_to_lds_b128 %0, %1, off"
               :: "v"(loff), "v"(gsrc)
               : "memory");
}
__device__ __forceinline__ void wait_async_le3() { asm volatile("s_wait_asynccnt 3" ::: "memory"); }
__device__ __forceinline__ void wait_async_le2() { asm volatile("s_wait_asynccnt 2" ::: "memory"); }
__device__ __forceinline__ void wait_async_le1() { asm volatile("s_wait_asynccnt 1" ::: "memory"); }
__device__ __forceinline__ void wait_async_le0() { asm volatile("s_wait_asynccnt 0" ::: "memory"); }

// ---- register-resident sorted top-5 ----
struct Top5 { float v0, v1, v2, v3, v4; int i0, i1, i2, i3, i4; };

__device__ __forceinline__ void top5_insert(Top5& t, float p, int idx) {
  if (p > t.v4) {
    t.v4 = p; t.i4 = idx;
    if (t.v4 > t.v3) { float v = t.v3; t.v3 = t.v4; t.v4 = v; int i = t.i3; t.i3 = t.i4; t.i4 = i; }
    if (t.v3 > t.v2) { float v = t.v2; t.v2 = t.v3; t.v3 = v; int i = t.i2; t.i2 = t.i3; t.i3 = i; }
    if (t.v2 > t.v1) { float v = t.v1; t.v1 = t.v2; t.v2 = v; int i = t.i1; t.i1 = t.i2; t.i2 = i; }
    if (t.v1 > t.v0) { float v = t.v0; t.v0 = t.v1; t.v1 = v; int i = t.i0; t.i0 = t.i1; t.i1 = i; }
  }
}

__device__ __forceinline__ float sigmoidf(float x) { return 1.0f / (1.0f + expf(-x)); }

__global__ __launch_bounds__(TPB)
void soft_mask_kernel(const int*   __restrict__ x_t,
                      const float* __restrict__ probs,
                      const float* __restrict__ emb,
                      const float* __restrict__ om_s,
                      const float* __restrict__ om_a,
                      const float* __restrict__ om_b,
                      float*       __restrict__ out) {
  __shared__ float4 sbuf[kDepth][TPB];   // 4-deep async ring (16 KB)
  __shared__ float  red_v[TPB];
  __shared__ int    red_i[TPB];
  __shared__ int    red_t[TPB];
  __shared__ float  s_ent;
  __shared__ float  s_lam;
  __shared__ float  s_w[kK];
  __shared__ int    s_idx[kK];
  __shared__ int    s_win;

  const int row = blockIdx.x;                  // 0 .. B*S-1
  const int tid = threadIdx.x;
  const float* prow = probs + (size_t)row * kV;

  // prologue: prefetch tiles 0..2 (3 outstanding per wave)
  #pragma unroll
  for (int pf = 0; pf < kDepth - 1; ++pf) {
    async_copy_b128(prow + pf * kChunk + tid * 4, &sbuf[pf][tid]);
  }

  float ent = 0.0f;
  Top5 t;
  t.v0 = t.v1 = t.v2 = t.v3 = t.v4 = -3.4028235e38f;
  t.i0 = t.i1 = t.i2 = t.i3 = t.i4 = 0;

  // ---- phase 1: stream probs row through LDS (4-deep async DMA pipeline) ----
  for (int it = 0; it < kIters; ++it) {
    const int nxt = it + (kDepth - 1);
    if (nxt < kIters) {
      async_copy_b128(prow + nxt * kChunk + tid * 4, &sbuf[nxt & (kDepth - 1)][tid]);
      wait_async_le3();                       // oldest tile (it) has landed
    } else {
      const int rem = kIters - 1 - it;        // tiles still outstanding after this one
      if (rem >= 2)      wait_async_le2();
      else if (rem == 1) wait_async_le1();
      else               wait_async_le0();
    }
    float4 p4 = sbuf[it & (kDepth - 1)][tid];
    const int base = it * kChunk + tid * 4;
    // probs are always normal-range floats: hardware v_log_f32 is plenty accurate
    ent = fmaf(p4.x, __logf(p4.x + 1e-10f), ent); top5_insert(t, p4.x, base + 0);
    ent = fmaf(p4.y, __logf(p4.y + 1e-10f), ent); top5_insert(t, p4.y, base + 1);
    ent = fmaf(p4.z, __logf(p4.z + 1e-10f), ent); top5_insert(t, p4.z, base + 2);
    ent = fmaf(p4.w, __logf(p4.w + 1e-10f), ent); top5_insert(t, p4.w, base + 3);
  }
  if (tid < kTail) {
    const int idx = kIters * kChunk + tid;
    const float p = prow[idx];
    ent = fmaf(p, __logf(p + 1e-10f), ent);
    top5_insert(t, p, idx);
  }

  // ---- entropy block-reduce ----
  red_v[tid] = ent;
  __syncthreads();
  for (int s = TPB / 2; s > 0; s >>= 1) {
    if (tid < s) red_v[tid] += red_v[tid + s];
    __syncthreads();
  }
  if (tid == 0) s_ent = red_v[0];   // s_ent = sum p*log(p+eps) = -entropy
  __syncthreads();

  // ---- 5 rounds of block argmax over each lane's sorted-list head ----
  int h = 0;
  for (int r = 0; r < kK; ++r) {
    float cv = -3.4028235e38f; int ci = 0;
    if      (h == 0) { cv = t.v0; ci = t.i0; }
    else if (h == 1) { cv = t.v1; ci = t.i1; }
    else if (h == 2) { cv = t.v2; ci = t.i2; }
    else if (h == 3) { cv = t.v3; ci = t.i3; }
    else if (h == 4) { cv = t.v4; ci = t.i4; }
    red_v[tid] = cv; red_i[tid] = ci; red_t[tid] = tid;
    __syncthreads();
    for (int s = TPB / 2; s > 0; s >>= 1) {
      if (tid < s && red_v[tid + s] > red_v[tid]) {
        red_v[tid] = red_v[tid + s];
        red_i[tid] = red_i[tid + s];
        red_t[tid] = red_t[tid + s];
      }
      __syncthreads();
    }
    if (tid == 0) { s_w[r] = red_v[0]; s_idx[r] = red_i[0]; s_win = red_t[0]; }
    __syncthreads();
    if (tid == s_win) ++h;          // winner pops its head
    __syncthreads();
  }

  // ---- lambda + normalized top-k weights ----
  if (tid == 0) {
    const float os = om_s[0], oa = om_a[0], ob = om_b[0];
    const float sp = (oa > 20.0f) ? oa : log1pf(expf(oa));   // softplus
    s_lam = sigmoidf(os) * sigmoidf(sp * (s_ent - ob));      // sp * (-H(p) - ob)
    const float wsum = s_w[0] + s_w[1] + s_w[2] + s_w[3] + s_w[4];
    const float inv  = 1.0f / wsum;
    for (int k = 0; k < kK; ++k) s_w[k] *= inv;
  }
  __syncthreads();

  // ---- phase 2: emit H=1024 output floats (one float4 per lane) ----
  const int token = x_t[row];                  // uniform per block -> no divergence
  const float4* e4 = reinterpret_cast<const float4*>(emb);
  float4* o4 = reinterpret_cast<float4*>(out);
  const int hq = kH / 4;                       // 256 float4 per row
  float4 res;
  if (token != kMask) {
    res = e4[(size_t)token * hq + tid];
  } else {
    const float lam = s_lam;
    const float c = 1.0f - lam;
    const float4 m = e4[(size_t)kMask * hq + tid];
    res.x = c * m.x; res.y = c * m.y; res.z = c * m.z; res.w = c * m.w;
    #pragma unroll
    for (int k = 0; k < kK; ++k) {
      const float wk = lam * s_w[k];
      const float4 ek = e4[(size_t)s_idx[k] * hq + tid];
      res.x += wk * ek.x; res.y += wk * ek.y; res.z += wk * ek.z; res.w += wk * ek.w;
    }
  }
  o4[(size_t)row * hq + tid] = res;
}

extern "C" void kernel_launch(void* const* d_in, const int* in_sizes, int n_in,
                              void* d_out, int out_size, void* d_ws, size_t ws_size,
                              hipStream_t stream) {
  const int*   x_t   = (const int*)  d_in[0];
  const float* probs = (const float*)d_in[1];
  const float* emb   = (const float*)d_in[2];
  const float* oms   = (const float*)d_in[3];
  const float* oma   = (const float*)d_in[4];
  const float* omb   = (const float*)d_in[5];
  float* out = (float*)d_out;
  const int rows = in_sizes[0];   // B*S = 4096
  soft_mask_kernel<<<dim3(rows), dim3(TPB), 0, stream>>>(x_t, probs, emb, oms, oma, omb, out);
  (void)n_in; (void)out_size; (void)d_ws; (void)ws_size;
}